// HealEncoding_77764677862011
// MI455X (gfx1250) — compile-verified
//
#include <hip/hip_runtime.h>

// HealEncoding: L=4 levels, F=2 features, B=1e6 points.
// Memory-bound gather + inverse-distance weighted 8-neighbor sum.
// params table (4*792*2 floats = 25,344 B) is staged into LDS via the CDNA5
// async-to-LDS path (global_load_async_to_lds_b128 / s_wait_asynccnt), then
// gathered with ds_load_b64; everything else is coalesced streaming.

#define NLEV   4
#define MAXSZ  792
#define FD     2
#define TABLE_ELEMS (NLEV * MAXSZ * FD)   // 6336 floats = 25,344 bytes LDS

#define AS1 __attribute__((address_space(1)))
#define AS3 __attribute__((address_space(3)))

typedef int v4i __attribute__((vector_size(16)));   // matches builtin param type

#if defined(__has_builtin)
#if __has_builtin(__builtin_amdgcn_global_load_async_to_lds_b128) && \
    __has_builtin(__builtin_amdgcn_s_wait_asynccnt)
#define USE_ASYNC_LDS 1
#endif
#if __has_builtin(__builtin_amdgcn_rsqf)
#define FAST_RSQ(x) __builtin_amdgcn_rsqf(x)      // raw v_rsq_f32, ~1 ulp
#endif
#endif
#ifndef USE_ASYNC_LDS
#define USE_ASYNC_LDS 0
#endif
#ifndef FAST_RSQ
#define FAST_RSQ(x) rsqrtf(x)
#endif

__launch_bounds__(256, 2)
__global__ void heal_encoding_kernel(const float* __restrict__ params,   // [4,792,2]
                                     const float* __restrict__ platlon,  // [4,B,2]
                                     const float* __restrict__ nlatlon,  // [4,8B,2]
                                     const int*   __restrict__ pidx,     // [4,B]
                                     const int*   __restrict__ nidx,     // [4,8,B]
                                     float*       __restrict__ out,      // [B,8] = out[b*8+f*4+l]
                                     int B) {
    __shared__ float s_par[TABLE_ELEMS];
    const int tid = threadIdx.x;

    // ---- Stage the parameter table into LDS (CDNA5 async copy, ASYNCcnt) ----
#if USE_ASYNC_LDS
    for (int i = tid * 4; i < TABLE_ELEMS; i += 256 * 4) {
        __builtin_amdgcn_global_load_async_to_lds_b128(
            (AS1 v4i*)(params + i), (AS3 v4i*)(&s_par[i]),
            /*imm offset*/0, /*cpol*/0);
    }
    __builtin_amdgcn_s_wait_asynccnt(0);
#else
    for (int i = tid; i < TABLE_ELEMS; i += 256) s_par[i] = params[i];
#endif
    __syncthreads();

    const int gsize = gridDim.x * blockDim.x;
    for (int b = blockIdx.x * blockDim.x + tid; b < B; b += gsize) {
        float acc[NLEV][FD];

#pragma unroll
        for (int l = 0; l < NLEV; ++l) {
            const size_t pb  = (size_t)l * (size_t)B + (size_t)b;
            const float2 mll = *(const float2*)(platlon + 2 * pb);  // (theta, phi)

            // residual: my_reps (pixel_index is never -1 in-domain; clamp for safety)
            int pi = pidx[pb];
            pi = (pi < 0) ? 0 : ((pi > MAXSZ - 1) ? (MAXSZ - 1) : pi);
            const float* pr = &s_par[(l * MAXSZ + pi) * FD];
            float a0 = pr[0];
            float a1 = pr[1];

#pragma unroll
            for (int n = 0; n < 8; ++n) {
                const size_t off = ((size_t)l * 8 + (size_t)n) * (size_t)B + (size_t)b;
                int ni = nidx[off];
                const float msk = (ni >= 0) ? 1.0f : 0.0f;   // missing-neighbor mask
                ni = (ni < 0) ? 0 : ni;                      // masked gather -> row 0 * 0

                const float2 nll = *(const float2*)(nlatlon + 2 * off);
                const float dt = nll.x - mll.x;
                const float dp = nll.y - mll.y;
                // w = 1/sqrt(dphi^2 + dtheta^2), masked  (single v_rsq_f32)
                const float w = msk * FAST_RSQ(fmaf(dp, dp, dt * dt));

                const float* nr = &s_par[(l * MAXSZ + ni) * FD];  // ds_load_b64 gather
                a0 = fmaf(nr[0], w, a0);
                a1 = fmaf(nr[1], w, a1);
            }
            acc[l][0] = a0;
            acc[l][1] = a1;
        }

        // out layout from reshape(4,-1).T.reshape(-1,8): out[b*8 + f*4 + l]
        float4* orow = (float4*)(out + (size_t)b * 8);
        orow[0] = make_float4(acc[0][0], acc[1][0], acc[2][0], acc[3][0]);  // f=0, l=0..3
        orow[1] = make_float4(acc[0][1], acc[1][1], acc[2][1], acc[3][1]);  // f=1, l=0..3

        // prefetch next grid-stride tile of the dominant streams (global_prefetch_b8)
        const int bn = b + gsize;
        if (bn < B) {
            __builtin_prefetch(nidx + (size_t)bn, 0, 1);
            __builtin_prefetch(nlatlon + 2 * (size_t)bn, 0, 1);
        }
    }
}

extern "C" void kernel_launch(void* const* d_in, const int* in_sizes, int n_in,
                              void* d_out, int out_size, void* d_ws, size_t ws_size,
                              hipStream_t stream) {
    (void)n_in; (void)out_size; (void)d_ws; (void)ws_size;
    const float* params  = (const float*)d_in[0];  // [4,792,2] f32
    const float* platlon = (const float*)d_in[1];  // [4,B,2]   f32
    const float* nlatlon = (const float*)d_in[2];  // [4,8B,2]  f32
    const int*   pidx    = (const int*)d_in[3];    // [4,B]     i32
    const int*   nidx    = (const int*)d_in[4];    // [4,8,B]   i32
    float*       out     = (float*)d_out;          // [B,8]     f32

    const int B = in_sizes[3] / NLEV;              // pixel_index has 4*B elements

    const int threads = 256;
    int blocks = (B + threads - 1) / threads;
    if (blocks > 1024) blocks = 1024;              // grid-stride; amortize LDS staging
    heal_encoding_kernel<<<dim3(blocks), dim3(threads), 0, stream>>>(
        params, platlon, nlatlon, pidx, nidx, out, B);
}